// RegionLoss_DIGER_30958124269944
// MI455X (gfx1250) — compile-verified
//
#include <hip/hip_runtime.h>
#include <stdint.h>

#define BB 32
#define AA 3
#define HH 76
#define WWD 76
#define MM 50
#define HWp (HH*WWD)            // 5776
#define NNr (AA*HWp)            // 17328
#define BNr (BB*NNr)            // 554496
#define ROWS_PB 256
#define NBLOCKS (BNr/ROWS_PB)   // 2166 exactly
#define EPSF 1e-7f

typedef unsigned int u32;
typedef __attribute__((ext_vector_type(4))) u32 v4u;
typedef __attribute__((ext_vector_type(8))) int v8i;
typedef __attribute__((ext_vector_type(4))) int v4i;

__device__ __forceinline__ float sigmoidf_(float x) { return 1.0f / (1.0f + expf(-x)); }

__global__ __launch_bounds__(256) void region_loss_main(
    const float* __restrict__ out_t,      // [B, A*8, H, W]
    const float* __restrict__ bbox_gt,    // [B, 4, 1, M]
    const float* __restrict__ conf_mask_g,// [B, N, M]
    const int*   __restrict__ cls_mask,   // [B, N]
    float* __restrict__ partials)         // [NBLOCKS, 4]
{
  __shared__ float s_mask[ROWS_PB * MM];  // 51200 B tile of conf_gt_mask
  __shared__ float s_gt[2 * 4 * MM];      // gt for the (up to) 2 batches in this block
  __shared__ float s_part[8][4];          // per-wave partials

  const int tid = threadIdx.x;
  const int blk = blockIdx.x;
  const long long r0 = (long long)blk * ROWS_PB;
  const int b0 = (int)(r0 / NNr);         // one scalar division per block
  const int n0 = (int)(r0 - (long long)b0 * NNr);

  // --- Issue TDM bulk copy (no wait yet): 256 contiguous mask rows (51200 B)
  // --- from HBM -> LDS; wave 0 only. DMA overlaps the decode work below.
  if (tid < 32) {
    unsigned long long ga = (unsigned long long)(uintptr_t)conf_mask_g
                          + (unsigned long long)r0 * (MM * 4ull);
    u32 lds_addr = (u32)(size_t)(&s_mask[0]);
    const u32 NDW = ROWS_PB * MM;         // 12800 dwords, fits tile_dim0 (16b)
    v4u g0; v8i g1; v4i g2; v4i g3; v8i g4;
    // D# group 0: count=1, lds_addr, global_addr[56:0], type=2
    g0[0] = 1u;
    g0[1] = lds_addr;
    g0[2] = (u32)(ga & 0xFFFFFFFFull);
    g0[3] = (u32)((ga >> 32) & 0x01FFFFFFull) | (2u << 30);
    // D# group 1: data_size=4B (code 2); tensor_dim0=NDW; tensor_dim1=1;
    // tile_dim0=NDW; tile_dim1=1; tile_dim2=0; dim0_stride=NDW; dim1_stride=0
    g1[0] = (int)(2u << 16);
    g1[1] = (int)((NDW & 0xFFFFu) << 16);
    g1[2] = (int)(((NDW >> 16) & 0xFFFFu) | (1u << 16));
    g1[3] = (int)((NDW & 0xFFFFu) << 16);
    g1[4] = 1;
    g1[5] = (int)NDW;
    g1[6] = 0;
    g1[7] = 0;
    g2[0] = 0; g2[1] = 0; g2[2] = 0; g2[3] = 0;
    g3[0] = 0; g3[1] = 0; g3[2] = 0; g3[3] = 0;
    g4[0] = 0; g4[1] = 0; g4[2] = 0; g4[3] = 0;
    g4[4] = 0; g4[5] = 0; g4[6] = 0; g4[7] = 0;
    __builtin_amdgcn_tensor_load_to_lds(g0, g1, g2, g3, g4, 0);
  }

  // Stage GT table for batches b0 and b0+1 (clamped at end of buffer).
  for (int i = tid; i < 2 * 4 * MM; i += 256) {
    long long gi = (long long)b0 * (4 * MM) + i;
    if (gi > (long long)BB * 4 * MM - 1) gi = (long long)BB * 4 * MM - 1;
    s_gt[i] = bbox_gt[gi];
  }

  // --- per-row indexing without 64-bit division: block spans <= 2 batches ---
  const int nt   = n0 + tid;
  const int wrap = (nt >= NNr) ? 1 : 0;
  const int b    = b0 + wrap;
  const int n    = nt - (wrap ? NNr : 0);
  const int a    = n / HWp;
  const int hw   = n % HWp;
  const int hrw  = hw / WWD;
  const int wcl  = hw % WWD;
  const float aw = (a == 0) ? 1.3f : ((a == 1) ? 3.2f : 5.0f);
  const float ah = (a == 0) ? 1.7f : ((a == 1) ? 4.1f : 6.2f);

  // Channel loads + decode + focal: independent of the mask tile -> overlaps TDM.
  const float* ch = out_t + ((size_t)b * (AA * 8) + (size_t)a * 8) * HWp + hw;
  float t0 = ch[0];
  float t1 = ch[(size_t)1 * HWp];
  float t2 = ch[(size_t)2 * HWp];
  float t3 = ch[(size_t)3 * HWp];
  float t4 = ch[(size_t)4 * HWp];
  float l0 = ch[(size_t)5 * HWp];
  float l1 = ch[(size_t)6 * HWp];
  float l2 = ch[(size_t)7 * HWp];

  float sx = sigmoidf_(t0), sy = sigmoidf_(t1), conf = sigmoidf_(t4);
  float cx = sx + (float)wcl;
  float cy = sy + (float)hrw;
  float pw = aw * expf(t2);
  float ph = ah * expf(t3);
  float px1 = cx - 0.5f * pw, py1 = cy - 0.5f * ph;
  float px2 = cx + 0.5f * pw, py2 = cy + 0.5f * ph;
  float parea = pw * ph;

  // focal loss (3 classes) — also independent of the mask tile
  int tc = cls_mask[r0 + tid];
  float mxl = fmaxf(l0, fmaxf(l1, l2));
  float lse = mxl + logf(expf(l0 - mxl) + expf(l1 - mxl) + expf(l2 - mxl));
  float lt = (tc == 0) ? l0 : ((tc == 1) ? l1 : l2);
  float lpt = lt - lse;
  float pt = expf(lpt);
  float om = 1.f - pt;
  float focal = -om * om * lpt;

  // --- now drain the DMA and make the tile visible to all waves ---
  if (tid < 32) {
    __builtin_amdgcn_s_wait_tensorcnt(0);
    asm volatile("" ::: "memory");
  }
  __syncthreads();

  float iou_max = 0.f, gtiou = 0.f, assigned = 0.f;
  float gbcx = 0.f, gbcy = 0.f, gbw = 0.f, gbh = 0.f;
  const int sel = wrap * (4 * MM);
  const float* mrow = &s_mask[tid * MM];

  for (int m = 0; m < MM; ++m) {
    float gcx = s_gt[sel + m];
    float gcy = s_gt[sel + MM + m];
    float gw  = s_gt[sel + 2 * MM + m];
    float gh  = s_gt[sel + 3 * MM + m];
    float gx1 = gcx - 0.5f * gw, gy1 = gcy - 0.5f * gh;
    float gx2 = gcx + 0.5f * gw, gy2 = gcy + 0.5f * gh;
    float iw = fmaxf(fminf(px2, gx2) - fmaxf(px1, gx1), 0.f);
    float ih = fmaxf(fminf(py2, gy2) - fmaxf(py1, gy1), 0.f);
    float inter = iw * ih;
    float uni = parea + gw * gh - inter;
    float iou = inter / (uni + EPSF);
    float mk = mrow[m];
    iou_max  = fmaxf(iou_max, iou);
    gtiou    = fmaxf(gtiou, iou * mk);
    assigned = fmaxf(assigned, mk);
    gbcx = fmaxf(gbcx, gcx * mk);
    gbcy = fmaxf(gbcy, gcy * mk);
    gbw  = fmaxf(gbw,  gw * mk);
    gbh  = fmaxf(gbh,  gh * mk);
  }

  // conf loss term (reference order: no_obj -> ignore -> obj)
  float cm = 1.0f;
  if (iou_max > 0.6f)  cm = 0.0f;
  if (assigned > 0.f)  cm = 5.0f;
  float cerr = conf * cm - gtiou * cm;
  float conf_sq = cerr * cerr;

  float valid = (gbcx > 0.f && gbcy > 0.f && gbw > 0.f && gbh > 0.f) ? 1.f : 0.f;

  // CIoU on normalized xyxy
  const float inv = 1.0f / 76.0f;
  float Px1 = px1 * inv, Py1 = py1 * inv, Px2 = px2 * inv, Py2 = py2 * inv;
  float Gx1 = (gbcx - 0.5f * gbw) * inv, Gy1 = (gbcy - 0.5f * gbh) * inv;
  float Gx2 = (gbcx + 0.5f * gbw) * inv, Gy2 = (gbcy + 0.5f * gbh) * inv;
  float iw2 = fmaxf(fminf(Px2, Gx2) - fmaxf(Px1, Gx1), 0.f);
  float ih2 = fmaxf(fminf(Py2, Gy2) - fmaxf(Py1, Gy1), 0.f);
  float inter2 = iw2 * ih2;
  float uni2 = (Px2 - Px1) * (Py2 - Py1) + (Gx2 - Gx1) * (Gy2 - Gy1) - inter2;
  float iou2 = inter2 / (uni2 + EPSF);
  float cwv = fmaxf(Px2, Gx2) - fminf(Px1, Gx1);
  float chv = fmaxf(Py2, Gy2) - fminf(Py1, Gy1);
  float c2 = cwv * cwv + chv * chv + EPSF;
  float dxs = Px1 + Px2 - Gx1 - Gx2, dys = Py1 + Py2 - Gy1 - Gy2;
  float rho2 = (dxs * dxs + dys * dys) * 0.25f;
  float dv = atanf((Gx2 - Gx1) / (Gy2 - Gy1 + EPSF)) - atanf((Px2 - Px1) / (Py2 - Py1 + EPSF));
  float v = 0.4052847345693511f * dv * dv;   // 4/pi^2
  float alpha = v / (1.f - iou2 + v + EPSF);
  float ciou = 1.f - iou2 + rho2 / c2 + alpha * v;

  // --- deterministic reduction: wave32 shuffle tree, then wave 0 partials ---
  float v0 = ciou * valid, v1 = valid, v2 = conf_sq, v3 = focal;
  for (int off = 16; off > 0; off >>= 1) {
    v0 += __shfl_down(v0, off, 32);
    v1 += __shfl_down(v1, off, 32);
    v2 += __shfl_down(v2, off, 32);
    v3 += __shfl_down(v3, off, 32);
  }
  const int wv = tid >> 5, ln = tid & 31;
  if (ln == 0) {
    s_part[wv][0] = v0; s_part[wv][1] = v1; s_part[wv][2] = v2; s_part[wv][3] = v3;
  }
  __syncthreads();
  if (tid == 0) {
    float a0 = 0.f, a1 = 0.f, a2 = 0.f, a3 = 0.f;
    for (int w = 0; w < 8; ++w) {
      a0 += s_part[w][0]; a1 += s_part[w][1]; a2 += s_part[w][2]; a3 += s_part[w][3];
    }
    float* p = partials + (size_t)blk * 4;
    p[0] = a0; p[1] = a1; p[2] = a2; p[3] = a3;
  }
}

__global__ __launch_bounds__(256) void region_loss_finish(
    const float* __restrict__ partials, float* __restrict__ out)
{
  __shared__ float s0[256], s1[256], s2[256], s3[256];
  int tid = threadIdx.x;
  float a0 = 0.f, a1 = 0.f, a2 = 0.f, a3 = 0.f;
  for (int i = tid; i < NBLOCKS; i += 256) {
    const float* p = partials + (size_t)i * 4;
    a0 += p[0]; a1 += p[1]; a2 += p[2]; a3 += p[3];
  }
  s0[tid] = a0; s1[tid] = a1; s2[tid] = a2; s3[tid] = a3;
  __syncthreads();
  for (int s = 128; s > 0; s >>= 1) {
    if (tid < s) {
      s0[tid] += s0[tid + s];
      s1[tid] += s1[tid + s];
      s2[tid] += s2[tid + s];
      s3[tid] += s3[tid + s];
    }
    __syncthreads();
  }
  if (tid == 0) {
    float sum_ciou = s0[0], sum_valid = s1[0], sum_conf = s2[0], sum_focal = s3[0];
    float xywh_loss = sum_ciou / fmaxf(sum_valid, 1.0f) * 15.0f;
    float conf_loss = (sum_conf / (float)BNr) * 10.0f;
    float cls_loss  = 1.0f * (sum_focal / (float)BNr) * 1.5f;
    out[0] = xywh_loss + conf_loss + cls_loss;
  }
}

extern "C" void kernel_launch(void* const* d_in, const int* in_sizes, int n_in,
                              void* d_out, int out_size, void* d_ws, size_t ws_size,
                              hipStream_t stream) {
  const float* out_t = (const float*)d_in[0];   // output [B,A*8,H,W]
  const float* bbox  = (const float*)d_in[1];   // bbox_gt [B,4,1,M]
  const float* cmask = (const float*)d_in[2];   // conf_gt_mask [B,N,M]
  // d_in[3] = xywh_mask : dead input, never read by the reference
  const int*   clsm  = (const int*)d_in[4];     // cls_mask [B,N]
  float* partials = (float*)d_ws;               // NBLOCKS*4 floats = 34656 B

  region_loss_main<<<NBLOCKS, 256, 0, stream>>>(out_t, bbox, cmask, clsm, partials);
  region_loss_finish<<<1, 256, 0, stream>>>(partials, (float*)d_out);
}